// SpectralConv2d_dse_9457517985876
// MI455X (gfx1250) — compile-verified
//
#include <hip/hip_runtime.h>
#include <hip/hip_bf16.h>

typedef float v2f __attribute__((ext_vector_type(2)));
typedef float v8f __attribute__((ext_vector_type(8)));

#define B_   8
#define C_   32
#define N_   4096
#define M1_  16
#define EMB_ 256
#define M_   992   // 2*M1*(2*M1-1)

// ---- workspace layout (float offsets) ----
#define WS_PHI_R   0
#define WS_PHI_I   (WS_PHI_R   + B_*256)
#define WS_XFT_R   (WS_PHI_I   + B_*256)                 // [b][m][c]
#define WS_XFT_I   (WS_XFT_R   + (size_t)B_*M_*C_)
#define WS_FLAT_R  (WS_XFT_I   + (size_t)B_*M_*C_)       // [b][c][512]
#define WS_FLAT_I  (WS_FLAT_R  + (size_t)B_*C_*512)
#define WS_FULL_R  (WS_FLAT_I  + (size_t)B_*C_*512)      // [b][c][992] (transposed)
#define WS_FULL_NI (WS_FULL_R  + (size_t)B_*C_*M_)       // negated imag

// Non-temporal streaming load of a WMMA f32 fragment (2 VGPRs).
// The V matrices are consumed exactly once -> NT hint keeps the reused
// operands (x, fullT) resident in L2 while 520 MB stream through.
static __device__ __forceinline__ v2f ldnt2(const float* p) {
    return __builtin_nontemporal_load((const v2f*)p);
}

// ---------------------------------------------------------------------------
// Kernel 1: phi[b,x,y] = sum_c A[x,y,c] * emb[b,c]   (complex, tiny)
// ---------------------------------------------------------------------------
__global__ __launch_bounds__(256)
void phi_kernel(const float* __restrict__ A_r, const float* __restrict__ A_i,
                const float* __restrict__ emb, float* __restrict__ ws) {
    const int b = blockIdx.x;
    const int t = threadIdx.x;                 // x*16 + y
    const float* e  = emb + (size_t)b * EMB_;
    const float* ar = A_r + (size_t)t * EMB_;
    const float* ai = A_i + (size_t)t * EMB_;
    float sr = 0.f, si = 0.f;
#pragma unroll 8
    for (int c = 0; c < EMB_; ++c) {
        float ev = e[c];
        sr += ar[c] * ev;
        si += ai[c] * ev;
    }
    ws[WS_PHI_R + b * 256 + t] = sr;
    ws[WS_PHI_I + b * 256 + t] = si;
}

// ---------------------------------------------------------------------------
// Kernel 2: forward NUFFT  x_ft[b,m,c] = sum_n Vf[b,m,n] * x[b,c,n]
// One block = one 16-row M tile of one batch. 4 waves: (real/imag) x (c-tile).
// WMMA f32 16x16x4, K swept over N=4096. Vf loads are non-temporal (stream).
// ---------------------------------------------------------------------------
__global__ __launch_bounds__(128)
void fwd_nufft_kernel(const float* __restrict__ x,
                      const float* __restrict__ Vf_r, const float* __restrict__ Vf_i,
                      float* __restrict__ ws) {
    const int b    = blockIdx.y;
    const int m0   = blockIdx.x * 16;
    const int lane = threadIdx.x & 31;
    const int wave = threadIdx.x >> 5;
    const int ri   = wave >> 1;              // 0 = real part, 1 = imag part
    const int c0   = (wave & 1) * 16;
    const int row  = lane & 15;              // M index for A, N(col) index for B
    const int hi   = lane >> 4;
    const int k2   = hi * 2;                 // per-lane K offset (A and B alike)

    const float* V  = ri ? Vf_i : Vf_r;
    const float* pA = V + ((size_t)b * M_ + (m0 + row)) * (size_t)N_ + k2;
    const float* pB = x + ((size_t)b * C_ + (c0 + row)) * (size_t)N_ + k2;

    v8f acc = {};
#pragma unroll 4
    for (int k = 0; k < N_; k += 4) {
        v2f a  = ldnt2(pA + k);              // streaming operand: NT
        v2f bb = *(const v2f*)(pB + k);      // reused operand: RT
        acc = __builtin_amdgcn_wmma_f32_16x16x4_f32(false, a, false, bb,
                                                    (short)0, acc, false, false);
    }
    float* dst = ws + (ri ? WS_XFT_I : WS_XFT_R) + (size_t)b * M_ * C_;
#pragma unroll
    for (int v = 0; v < 8; ++v)
        dst[(size_t)(m0 + v + 8 * hi) * C_ + c0 + row] = acc[v];
}

// ---------------------------------------------------------------------------
// Kernel 3: per-mode complex channel mixing + phase modulation -> flat[b,c,512]
// Block = (b, x*16+y); threads 0..31 do "top" (w1), 32..63 do "bot" (w2).
// ---------------------------------------------------------------------------
__global__ __launch_bounds__(64)
void mode_mix_kernel(const float* __restrict__ w1_r, const float* __restrict__ w1_i,
                     const float* __restrict__ w2_r, const float* __restrict__ w2_i,
                     float* __restrict__ ws) {
    const int b   = blockIdx.y;
    const int xy  = blockIdx.x;              // 0..255
    const int xm  = xy >> 4, y = xy & 15;
    const int tid = threadIdx.x;
    const int bot = tid >> 5;
    const int o   = tid & 31;
    const int m_in = (bot ? (xm + 16) : xm) * 31 + y;   // mode index in [0,992)
    const float* wr = bot ? w2_r : w1_r;
    const float* wi = bot ? w2_i : w1_i;
    const float* xr = ws + WS_XFT_R + ((size_t)b * M_ + m_in) * C_;
    const float* xi = ws + WS_XFT_I + ((size_t)b * M_ + m_in) * C_;
    float sr = 0.f, si = 0.f;
#pragma unroll 4
    for (int i = 0; i < C_; ++i) {
        float wrv = wr[(((size_t)i * C_ + o) * M1_ + xm) * M1_ + y];
        float wiv = wi[(((size_t)i * C_ + o) * M1_ + xm) * M1_ + y];
        float ar = xr[i], ai = xi[i];
        sr += ar * wrv - ai * wiv;
        si += ar * wiv + ai * wrv;
    }
    const float pr = ws[WS_PHI_R + b * 256 + xm * 16 + y];
    const float pi = ws[WS_PHI_I + b * 256 + xm * 16 + y];
    const float tr = sr * pr - si * pi;
    const float ti = sr * pi + si * pr;
    const int   X  = bot ? (xm + 16) : xm;
    const size_t fo = (size_t)b * C_ * 512 + (size_t)o * 512 + X * 16 + y;
    ws[WS_FLAT_R + fo] = tr;
    ws[WS_FLAT_I + fo] = ti;
}

// ---------------------------------------------------------------------------
// Kernel 4: conjugate-symmetric completion -> fullT[b][c][992], imag NEGATED
//   m <  512: full[b,m,c] = flat[b,c,m]
//   m >= 512: full[b,m,c] = conj(flat[b,31-c,1023-m])
// ---------------------------------------------------------------------------
__global__ __launch_bounds__(256)
void build_full_kernel(float* __restrict__ ws) {
    const int b   = blockIdx.y;
    const int idx = blockIdx.x * 256 + threadIdx.x;   // 0 .. 992*32-1
    if (idx >= M_ * C_) return;
    const int m = idx >> 5, c = idx & 31;
    float fr, fi;
    const size_t fb = (size_t)b * C_ * 512;
    if (m < 512) {
        fr =  ws[WS_FLAT_R + fb + (size_t)c * 512 + m];
        fi = -ws[WS_FLAT_I + fb + (size_t)c * 512 + m];        // -imag
    } else {
        const int sm = 1023 - m;
        const int sc = 31 - c;
        fr = ws[WS_FLAT_R + fb + (size_t)sc * 512 + sm];
        fi = ws[WS_FLAT_I + fb + (size_t)sc * 512 + sm];       // conj then negate -> +
    }
    ws[WS_FULL_R  + ((size_t)b * C_ + c) * M_ + m] = fr;
    ws[WS_FULL_NI + ((size_t)b * C_ + c) * M_ + m] = fi;
}

// ---------------------------------------------------------------------------
// Kernel 5: inverse NUFFT  y[b,c,n] = (2/N) * sum_m (ViR*fullR + ViI*(-fullI))
// 8 waves/block: 4 n-tiles x 2 c-tiles. Two chained WMMAs per K step.
// Vi loads are non-temporal (stream); fullT loads stay RT (reused).
// ---------------------------------------------------------------------------
__global__ __launch_bounds__(256)
void inv_nufft_kernel(const float* __restrict__ Vi_r, const float* __restrict__ Vi_i,
                      const float* __restrict__ ws, float* __restrict__ out) {
    const int b    = blockIdx.y;
    const int lane = threadIdx.x & 31;
    const int wave = threadIdx.x >> 5;                 // 0..7
    const int n0   = (blockIdx.x * 4 + (wave >> 1)) * 16;
    const int c0   = (wave & 1) * 16;
    const int row  = lane & 15;
    const int hi   = lane >> 4;
    const int k2   = hi * 2;

    const float* pAr = Vi_r + ((size_t)b * N_ + (n0 + row)) * (size_t)M_ + k2;
    const float* pAi = Vi_i + ((size_t)b * N_ + (n0 + row)) * (size_t)M_ + k2;
    const float* pBr = ws + WS_FULL_R  + ((size_t)b * C_ + (c0 + row)) * M_ + k2;
    const float* pBi = ws + WS_FULL_NI + ((size_t)b * C_ + (c0 + row)) * M_ + k2;

    v8f acc = {};
#pragma unroll 4
    for (int k = 0; k < M_; k += 4) {
        v2f ar = ldnt2(pAr + k);
        v2f br = *(const v2f*)(pBr + k);
        acc = __builtin_amdgcn_wmma_f32_16x16x4_f32(false, ar, false, br,
                                                    (short)0, acc, false, false);
        v2f ai = ldnt2(pAi + k);
        v2f bi = *(const v2f*)(pBi + k);
        acc = __builtin_amdgcn_wmma_f32_16x16x4_f32(false, ai, false, bi,
                                                    (short)0, acc, false, false);
    }
    const float scale = 2.0f / (float)N_;
#pragma unroll
    for (int v = 0; v < 8; ++v) {
        float r = acc[v] * scale;
        __builtin_nontemporal_store(
            r, out + ((size_t)b * C_ + (c0 + row)) * (size_t)N_ + n0 + v + 8 * hi);
    }
}

// ---------------------------------------------------------------------------
extern "C" void kernel_launch(void* const* d_in, const int* in_sizes, int n_in,
                              void* d_out, int out_size, void* d_ws, size_t ws_size,
                              hipStream_t stream) {
    const float* x      = (const float*)d_in[0];
    const float* emb    = (const float*)d_in[1];
    const float* Vf_r   = (const float*)d_in[2];
    const float* Vf_i   = (const float*)d_in[3];
    const float* Vi_r   = (const float*)d_in[4];
    const float* Vi_i   = (const float*)d_in[5];
    const float* w1_r   = (const float*)d_in[6];
    const float* w1_i   = (const float*)d_in[7];
    const float* w2_r   = (const float*)d_in[8];
    const float* w2_i   = (const float*)d_in[9];
    const float* A_real = (const float*)d_in[10];
    const float* A_imag = (const float*)d_in[11];
    float* out = (float*)d_out;
    float* ws  = (float*)d_ws;

    // 1) phase modulation phi[b,x,y]
    phi_kernel<<<dim3(B_), 256, 0, stream>>>(A_real, A_imag, emb, ws);

    // 2) forward NUFFT: 62 M-tiles x 8 batches, 4 waves/block (r/i x c-tile)
    fwd_nufft_kernel<<<dim3(M_ / 16, B_), 128, 0, stream>>>(x, Vf_r, Vf_i, ws);

    // 3) per-mode complex channel mixing + phase
    mode_mix_kernel<<<dim3(256, B_), 64, 0, stream>>>(w1_r, w1_i, w2_r, w2_i, ws);

    // 4) conjugate-symmetric completion (transposed, imag negated)
    build_full_kernel<<<dim3((M_ * C_) / 256, B_), 256, 0, stream>>>(ws);

    // 5) inverse NUFFT: 64 blocks x 8 batches, 8 waves/block
    inv_nufft_kernel<<<dim3(N_ / 64, B_), 256, 0, stream>>>(Vi_r, Vi_i, ws, out);
}